// ShawRelativePositionalEmbedding_22686017258312
// MI455X (gfx1250) — compile-verified
//
#include <hip/hip_runtime.h>
#include <hip/hip_bf16.h>

typedef __attribute__((ext_vector_type(2))) float v2f;
typedef __attribute__((ext_vector_type(8))) float v8f;

#define N_SEQ   2048
#define DIM     64
#define MAXLEN  2048
#define BH      32            // 2 * 16 batch-heads

// One block (8 waves) per query position n.
// Each wave computes 16 r-tiles of 16 columns; per tile it produces a 32x16
// output chunk via two chained fp32 WMMA accumulators sharing one B operand.
// Scale (1/sqrt(64) = 0.125, a power of two) is folded into A bit-exactly.
__global__ __launch_bounds__(256) void shaw_relpos_wmma_f32(
    const float* __restrict__ q,    // [32, 2048, 64]  (bh, n, d)
    const float* __restrict__ emb,  // [4097, 64]
    float* __restrict__ out)        // [32, 2048, 2048] (bh, n, r)
{
    const int n    = blockIdx.x;
    const int tid  = threadIdx.x;
    const int wave = tid >> 5;
    const int lane = tid & 31;
    const int m    = lane & 15;           // M row (A) / N col (B,C,D) within tile
    const int koff = (lane >> 4) << 1;    // 0 or 2: which K-pair this half-wave holds

    const float scale = 0.125f;  // 64^-0.5, exact power-of-two scale

    // ---- Load A once: Q_n rows for both 16-row M-tiles, K=64 as 16 chunks of 4 ----
    // A 16x4 f32 layout: lane L holds M=L%16, floats K = 2*(L/16), 2*(L/16)+1.
    v2f a0[16], a1[16];
    const float* qn = q + (size_t)n * DIM;
#pragma unroll
    for (int s = 0; s < 16; ++s) {
        const int k = s * 4 + koff;
        v2f t0 = *(const v2f*)(qn + (size_t)m        * (N_SEQ * DIM) + k);
        v2f t1 = *(const v2f*)(qn + (size_t)(m + 16) * (N_SEQ * DIM) + k);
        a0[s] = t0 * scale;   // exact: exponent shift only
        a1[s] = t1 * scale;
    }

    const int rowadd = (lane >> 4) << 3;              // D layout: lanes 16-31 hold M+8
    const size_t bh_stride = (size_t)N_SEQ * N_SEQ;

    // ---- r-tiles: wave w handles tiles w, w+8, ..., w+120 (all waves: 16 iters, no divergence) ----
    for (int rt = wave; rt < N_SEQ / 16; rt += 8) {
        const int r0 = rt * 16;
        // B column j (r = r0+j) is emb row (n - r + MAXLEN); j == lane&15 == m.
        // B 4x16 f32 layout mirrors A: lane L holds N=L%16, K-pair 2*(L/16).
        const float* ebase = emb + (size_t)(n + MAXLEN - r0 - m) * DIM + koff;

        v8f c0 = {0.f, 0.f, 0.f, 0.f, 0.f, 0.f, 0.f, 0.f};
        v8f c1 = {0.f, 0.f, 0.f, 0.f, 0.f, 0.f, 0.f, 0.f};
#pragma unroll
        for (int s = 0; s < 16; ++s) {
            const v2f b = *(const v2f*)(ebase + s * 4);
            // 8 args: (neg_a, A, neg_b, B, c_mod, C, reuse_a, reuse_b)
            c0 = __builtin_amdgcn_wmma_f32_16x16x4_f32(
                    false, a0[s], false, b, (short)0, c0, false, false);
            c1 = __builtin_amdgcn_wmma_f32_16x16x4_f32(
                    false, a1[s], false, b, (short)0, c1, false, false);
        }

        // ---- Store (non-temporal; output is write-once streaming 512MB):
        //      D VGPR v -> (M = v + rowadd, N = m). out[(n + M) * N_SEQ * N_SEQ + r]. ----
        const size_t base = (size_t)n * N_SEQ * N_SEQ + (size_t)r0 + (size_t)m;
        float* o0 = out + base;
#pragma unroll
        for (int v = 0; v < 8; ++v) {
            const int row = v + rowadd;
            __builtin_nontemporal_store(c0[v], o0 + (size_t)row        * bh_stride);
            __builtin_nontemporal_store(c1[v], o0 + (size_t)(row + 16) * bh_stride);
        }
    }
}

extern "C" void kernel_launch(void* const* d_in, const int* in_sizes, int n_in,
                              void* d_out, int out_size, void* d_ws, size_t ws_size,
                              hipStream_t stream) {
    (void)in_sizes; (void)n_in; (void)out_size; (void)d_ws; (void)ws_size;
    const float* q   = (const float*)d_in[0];   // [2,16,2048,64] -> [32,2048,64]
    // d_in[1] (k) is unused by the reference forward.
    const float* emb = (const float*)d_in[2];   // [4097, 64]
    float* out = (float*)d_out;                 // [2,16,2048,2048] -> [32,2048,2048]

    dim3 grid(N_SEQ);   // one block per n
    dim3 block(256);    // 8 waves (wave32)
    shaw_relpos_wmma_f32<<<grid, block, 0, stream>>>(q, emb, out);
}